// WindowAttention_33672543600796
// MI455X (gfx1250) — compile-verified
//
#include <hip/hip_runtime.h>
#include <hip/hip_bf16.h>

typedef float v2f __attribute__((ext_vector_type(2)));
typedef float v8f __attribute__((ext_vector_type(8)));

#define NTOK 343      // tokens per window (7*7*7)
#define NPAD 352      // padded to 22 tiles of 16
#define HEADS 6
#define HD 32
#define DIMC 192
#define BATCH 256
#define ATTN_SCALE 0.17677669529663687f  // 32^-0.5
#define SSTRIDE 353   // LDS score row stride (odd -> conflict-free column reads)

// relative position index, matching the numpy meshgrid('xy') flattening:
// cf0(f) = (f/7)%7, cf1(f) = f/49, cf2(f) = f%7
__device__ __forceinline__ int rp_index(int q, int k) {
  int q0 = (q / 7) % 7, q1 = q / 49, q2 = q % 7;
  int k0 = (k / 7) % 7, k1 = k / 49, k2 = k % 7;
  return (q0 - k0 + 6) * 169 + (q1 - k1 + 6) * 13 + (q2 - k2 + 6);
}

// ---------------------------------------------------------------------------
// Kernel 1: QKV projection  Y = X @ Wqkv, scattered to Q(scaled)/K/V
// X: [B*N, 192], W: [192, 576]; 16x16 output tile per wave, f32 WMMA K=4 steps
// ---------------------------------------------------------------------------
__global__ __launch_bounds__(128) void qkv_kernel(
    const float* __restrict__ X, const float* __restrict__ W,
    float* __restrict__ Q, float* __restrict__ K, float* __restrict__ V) {
  const int t = threadIdx.x;
  const int wid = t >> 5, lane = t & 31, half = lane >> 4, l16 = lane & 15;
  const int row0 = blockIdx.x * 16;
  const int col0 = (blockIdx.y * 4 + wid) * 16;  // 36 col tiles total
  const float* xr = X + (size_t)(row0 + l16) * DIMC;

  v8f acc = {};
  for (int kk = 0; kk < DIMC; kk += 4) {
    const int kc = kk + 2 * half;
    v2f a = { xr[kc], xr[kc + 1] };                               // A 16x4
    v2f b = { W[(size_t)kc * 576 + col0 + l16],                   // B 4x16
              W[(size_t)(kc + 1) * 576 + col0 + l16] };
    acc = __builtin_amdgcn_wmma_f32_16x16x4_f32(false, a, false, b,
                                                (short)0, acc, false, false);
  }

  const int which = col0 / DIMC;          // 0=q 1=k 2=v
  const int head  = (col0 % DIMC) / HD;
  const int hd0   = col0 % HD;            // 0 or 16
  float* dst = (which == 0) ? Q : (which == 1) ? K : V;
  const float sc = (which == 0) ? ATTN_SCALE : 1.0f;
#pragma unroll
  for (int i = 0; i < 8; ++i) {
    const int r = row0 + i + 8 * half;    // < B*N exactly (87808 = 5488*16)
    const int b = r / NTOK, n = r % NTOK;
    dst[(((size_t)b * HEADS + head) * NTOK + n) * HD + hd0 + l16] = acc[i] * sc;
  }
}

// ---------------------------------------------------------------------------
// Kernel 2: fused attention per (b, h, 16-query tile), 4 wave32 per block.
// Scores kept in LDS (16 x 352); softmax block-wide; P@V with LDS reduction.
// ---------------------------------------------------------------------------
__global__ __launch_bounds__(128) void attn_kernel(
    const float* __restrict__ Q, const float* __restrict__ K,
    const float* __restrict__ V, const float* __restrict__ mask,
    const float* __restrict__ rpb, float* __restrict__ AO) {
  __shared__ float sS[16 * SSTRIDE];
  __shared__ float sRed[16 * 8];
  __shared__ float sRowMax[16];
  __shared__ float sRowInv[16];
  __shared__ float sPart[2 * 16 * 16];

  const int t = threadIdx.x;
  const int wid = t >> 5, lane = t & 31, half = lane >> 4, l16 = lane & 15;
  const int q0 = blockIdx.x * 16;
  const int h  = blockIdx.y;
  const int b  = blockIdx.z;
  const size_t base = ((size_t)b * HEADS + h) * (size_t)NTOK * HD;
  const float* Qb = Q + base;
  const float* Kb = K + base;
  const float* Vb = V + base;

  // ---- phase 1: scores = q @ k^T + bias + mask, tiles split across waves ----
  float qa0[8], qa1[8];
  const int qrow = q0 + l16;
  const bool qok = qrow < NTOK;
#pragma unroll
  for (int s = 0; s < 8; ++s) {
    const int d = s * 4 + 2 * half;
    qa0[s] = qok ? Qb[qrow * HD + d]     : 0.f;
    qa1[s] = qok ? Qb[qrow * HD + d + 1] : 0.f;
  }
  const float* mrow = mask + (size_t)(b & 63) * NTOK * NTOK;
  for (int kt = wid; kt < NPAD / 16; kt += 4) {
    const int k0 = kt * 16;
    const int krow = k0 + l16;
    const bool kok = krow < NTOK;
    v8f acc = {};
#pragma unroll
    for (int s = 0; s < 8; ++s) {
      const int d = s * 4 + 2 * half;
      v2f a = { qa0[s], qa1[s] };
      v2f bf = { kok ? Kb[krow * HD + d]     : 0.f,   // B = k^T (d x key)
                 kok ? Kb[krow * HD + d + 1] : 0.f };
      acc = __builtin_amdgcn_wmma_f32_16x16x4_f32(false, a, false, bf,
                                                  (short)0, acc, false, false);
    }
#pragma unroll
    for (int i = 0; i < 8; ++i) {
      const int q = q0 + i + 8 * half;
      const int k = k0 + l16;
      float v = acc[i];
      if (q < NTOK && k < NTOK)
        v += rpb[rp_index(q, k) * HEADS + h] + mrow[q * NTOK + k];
      else
        v = -1e30f;
      sS[(i + 8 * half) * SSTRIDE + k] = v;
    }
  }
  __syncthreads();

  // ---- phase 2: softmax (store unnormalized exp; keep 1/rowsum) ----
  {
    const int r = t >> 3, sg = t & 7;
    float m = -1e30f;
    for (int k = sg; k < NTOK; k += 8) m = fmaxf(m, sS[r * SSTRIDE + k]);
    sRed[r * 8 + sg] = m;
    __syncthreads();
    if (sg == 0) {
      float mm = sRed[r * 8];
#pragma unroll
      for (int j = 1; j < 8; ++j) mm = fmaxf(mm, sRed[r * 8 + j]);
      sRowMax[r] = mm;
    }
    __syncthreads();
    const float rm = sRowMax[r];
    float sum = 0.f;
    for (int k = sg; k < NPAD; k += 8) {
      float e = 0.f;
      if (k < NTOK) { e = __expf(sS[r * SSTRIDE + k] - rm); sum += e; }
      sS[r * SSTRIDE + k] = e;  // zero the 9 padded columns
    }
    sRed[r * 8 + sg] = sum;
    __syncthreads();
    if (sg == 0) {
      float ss = 0.f;
#pragma unroll
      for (int j = 0; j < 8; ++j) ss += sRed[r * 8 + j];
      sRowInv[r] = 1.0f / ss;
    }
  }
  __syncthreads();

  // ---- phase 3: out = P @ V ; waves = (d-tile, k-half), reduce via LDS ----
  const int dt = wid & 1;
  const int kh = wid >> 1;
  const int d0 = dt * 16;
  v8f acc = {};
  for (int kk = kh * (NPAD / 2); kk < kh * (NPAD / 2) + NPAD / 2; kk += 4) {
    const int kc = kk + 2 * half;
    v2f a = { sS[l16 * SSTRIDE + kc], sS[l16 * SSTRIDE + kc + 1] };
    v2f bf = { (kc < NTOK)     ? Vb[(size_t)kc * HD + d0 + l16]       : 0.f,
               (kc + 1 < NTOK) ? Vb[(size_t)(kc + 1) * HD + d0 + l16] : 0.f };
    acc = __builtin_amdgcn_wmma_f32_16x16x4_f32(false, a, false, bf,
                                                (short)0, acc, false, false);
  }
  if (kh == 1) {
#pragma unroll
    for (int i = 0; i < 8; ++i)
      sPart[dt * 256 + (i + 8 * half) * 16 + l16] = acc[i];
  }
  __syncthreads();
  if (kh == 0) {
#pragma unroll
    for (int i = 0; i < 8; ++i) {
      const int q = q0 + i + 8 * half;
      if (q < NTOK) {
        const float v = (acc[i] + sPart[dt * 256 + (i + 8 * half) * 16 + l16])
                        * sRowInv[i + 8 * half];
        AO[((size_t)b * NTOK + q) * DIMC + h * HD + d0 + l16] = v;
      }
    }
  }
}

// ---------------------------------------------------------------------------
// Kernel 3: output projection  out = AO @ proj_w + proj_b
// ---------------------------------------------------------------------------
__global__ __launch_bounds__(128) void proj_kernel(
    const float* __restrict__ AO, const float* __restrict__ W,
    const float* __restrict__ bias, float* __restrict__ out) {
  const int t = threadIdx.x;
  const int wid = t >> 5, lane = t & 31, half = lane >> 4, l16 = lane & 15;
  const int row0 = blockIdx.x * 16;
  const int col0 = (blockIdx.y * 4 + wid) * 16;  // 12 col tiles
  const float* ar = AO + (size_t)(row0 + l16) * DIMC;

  v8f acc = {};
  for (int kk = 0; kk < DIMC; kk += 4) {
    const int kc = kk + 2 * half;
    v2f a = { ar[kc], ar[kc + 1] };
    v2f b = { W[(size_t)kc * DIMC + col0 + l16],
              W[(size_t)(kc + 1) * DIMC + col0 + l16] };
    acc = __builtin_amdgcn_wmma_f32_16x16x4_f32(false, a, false, b,
                                                (short)0, acc, false, false);
  }
  const float bv = bias[col0 + l16];
#pragma unroll
  for (int i = 0; i < 8; ++i) {
    const int r = row0 + i + 8 * half;
    out[(size_t)r * DIMC + col0 + l16] = acc[i] + bv;
  }
}

extern "C" void kernel_launch(void* const* d_in, const int* in_sizes, int n_in,
                              void* d_out, int out_size, void* d_ws, size_t ws_size,
                              hipStream_t stream) {
  const float* x      = (const float*)d_in[0];
  const float* mask   = (const float*)d_in[1];
  const float* qkv_w  = (const float*)d_in[2];
  const float* rpb    = (const float*)d_in[3];
  const float* proj_w = (const float*)d_in[4];
  const float* proj_b = (const float*)d_in[5];
  float* out = (float*)d_out;

  const size_t qkvElems = (size_t)BATCH * HEADS * NTOK * HD;  // 16,859,136
  float* Q  = (float*)d_ws;
  float* K  = Q + qkvElems;
  float* V  = K + qkvElems;
  float* AO = V + qkvElems;  // [B, N, 192]

  const int rowTiles = (BATCH * NTOK) / 16;  // 5488 (exact)
  qkv_kernel <<<dim3(rowTiles, 9),            128, 0, stream>>>(x, qkv_w, Q, K, V);
  attn_kernel<<<dim3(NPAD / 16, HEADS, BATCH), 128, 0, stream>>>(Q, K, V, mask, rpb, AO);
  proj_kernel<<<dim3(rowTiles, 3),            128, 0, stream>>>(AO, proj_w, proj_b, out);
}